// GNNLayer_48369921688359
// MI455X (gfx1250) — compile-verified
//
#include <hip/hip_runtime.h>
#include <hip/hip_bf16.h>
#include <math.h>

typedef __attribute__((ext_vector_type(2))) float v2f;
typedef __attribute__((ext_vector_type(8))) float v8f;

#define C64 64
#define NEG_SLOPE 0.2f
#define BN_EPS 1e-5f

// ---------------- order-preserving float<->uint encoding for atomicMax ------
__device__ __forceinline__ unsigned f2ord(float f) {
    unsigned u = __float_as_uint(f);
    return (u & 0x80000000u) ? ~u : (u | 0x80000000u);
}
__device__ __forceinline__ float ord2f(unsigned e) {
    unsigned u = (e & 0x80000000u) ? (e & 0x7FFFFFFFu) : ~e;
    return __uint_as_float(u);
}
// encoded -inf: f2ord(0xFF800000) = ~0xFF800000 = 0x007FFFFF
#define ORD_NEG_INF 0x007FFFFFu

// ---------------- kernel 0: init scratch + output accumulators --------------
__global__ void k_init(float* __restrict__ out, float* __restrict__ denom,
                       unsigned* __restrict__ amaxU, float* __restrict__ sums,
                       int n) {
    int i = blockIdx.x * blockDim.x + threadIdx.x;
    int total = n * C64;
    if (i < total) out[i] = 0.0f;
    if (i < n) { denom[i] = 0.0f; amaxU[i] = ORD_NEG_INF; }
    if (i < 128) sums[i] = 0.0f;
}

// ---------------- kernel 1: xl = x @ W^T via V_WMMA_F32_16X16X4_F32 ---------
// One wave computes a 16-row x 64-col tile. 4 accumulators (col tiles of 16),
// 16 K-steps of 4. W^T staged in LDS (WT[k*64+n] = W[n*64+k]).
// Fragment mapping per ISA 7.12.2:
//   A 16x4:  lane l, half=l>>4, lidx=l&15: row = row0+lidx,
//            a.x = A[row][k0+2*half], a.y = A[row][k0+2*half+1]
//   B 4x16:  col = col0+lidx, b.x = B[k0+2*half][col], b.y = B[k0+2*half+1][col]
//   C/D 16x16: acc[r] -> row = row0 + r + 8*half, col = col0 + lidx
__global__ __launch_bounds__(128) void k_gemm_xl(const float* __restrict__ x,
                                                 const float* __restrict__ W,
                                                 float* __restrict__ xl, int n) {
    __shared__ float WT[C64 * C64];
    for (int i = threadIdx.x; i < C64 * C64; i += 128) {
        int nn = i >> 6, k = i & 63;
        WT[k * C64 + nn] = W[nn * C64 + k];
    }
    __syncthreads();

    int wave = threadIdx.x >> 5;
    int lane = threadIdx.x & 31;
    int lidx = lane & 15;
    int half = lane >> 4;
    int row0 = (blockIdx.x * 4 + wave) * 16;
    if (row0 >= n) return;   // uniform per wave: EXEC stays all-1s for WMMA

    v8f acc0 = {}, acc1 = {}, acc2 = {}, acc3 = {};
    const float* arow = x + (size_t)(row0 + lidx) * C64;

    #pragma unroll
    for (int k0 = 0; k0 < C64; k0 += 4) {
        v2f a;
        a.x = arow[k0 + 2 * half];
        a.y = arow[k0 + 2 * half + 1];
        const float* bk0 = &WT[(k0 + 2 * half) * C64];
        const float* bk1 = &WT[(k0 + 2 * half + 1) * C64];
        v2f b0, b1, b2, b3;
        b0.x = bk0[lidx +  0]; b0.y = bk1[lidx +  0];
        b1.x = bk0[lidx + 16]; b1.y = bk1[lidx + 16];
        b2.x = bk0[lidx + 32]; b2.y = bk1[lidx + 32];
        b3.x = bk0[lidx + 48]; b3.y = bk1[lidx + 48];
        acc0 = __builtin_amdgcn_wmma_f32_16x16x4_f32(false, a, false, b0, (short)0, acc0, false, false);
        acc1 = __builtin_amdgcn_wmma_f32_16x16x4_f32(false, a, false, b1, (short)0, acc1, false, false);
        acc2 = __builtin_amdgcn_wmma_f32_16x16x4_f32(false, a, false, b2, (short)0, acc2, false, false);
        acc3 = __builtin_amdgcn_wmma_f32_16x16x4_f32(false, a, false, b3, (short)0, acc3, false, false);
    }

    #pragma unroll
    for (int r = 0; r < 8; ++r) {
        size_t o = (size_t)(row0 + r + 8 * half) * C64 + lidx;
        xl[o +  0] = acc0[r];
        xl[o + 16] = acc1[r];
        xl[o + 32] = acc2[r];
        xl[o + 48] = acc3[r];
    }
}

// ---------------- kernel 2: per-node attention scalars ----------------------
// ai[n] = <xl[n],att_i> + <emb[n],att_em_i>;  aj likewise. Wave-per-node.
__global__ __launch_bounds__(256) void k_node_att(const float* __restrict__ xl,
                                                  const float* __restrict__ emb,
                                                  const float* __restrict__ att_i,
                                                  const float* __restrict__ att_j,
                                                  const float* __restrict__ att_em_i,
                                                  const float* __restrict__ att_em_j,
                                                  float* __restrict__ ai,
                                                  float* __restrict__ aj, int n) {
    int node = blockIdx.x * 8 + (threadIdx.x >> 5);
    int lane = threadIdx.x & 31;
    if (node >= n) return;
    const float* xr = xl + (size_t)node * C64;
    const float* er = emb + (size_t)node * C64;
    float x0 = xr[lane], x1 = xr[lane + 32];
    float e0 = er[lane], e1 = er[lane + 32];
    float si = x0 * att_i[lane] + x1 * att_i[lane + 32]
             + e0 * att_em_i[lane] + e1 * att_em_i[lane + 32];
    float sj = x0 * att_j[lane] + x1 * att_j[lane + 32]
             + e0 * att_em_j[lane] + e1 * att_em_j[lane + 32];
    #pragma unroll
    for (int off = 16; off > 0; off >>= 1) {
        si += __shfl_down(si, off, 32);
        sj += __shfl_down(sj, off, 32);
    }
    if (lane == 0) { ai[node] = si; aj[node] = sj; }
}

// ---------------- kernel 3: edge logits + segment max -----------------------
__global__ void k_edge_logit(const int* __restrict__ src, const int* __restrict__ dst,
                             const float* __restrict__ ai, const float* __restrict__ aj,
                             float* __restrict__ abuf, unsigned* __restrict__ amaxU,
                             int E, int n) {
    int e = blockIdx.x * blockDim.x + threadIdx.x;
    if (e >= E + n) return;
    int s = (e < E) ? src[e] : (e - E);
    int d = (e < E) ? dst[e] : (e - E);
    float a = ai[d] + aj[s];
    a = (a >= 0.0f) ? a : NEG_SLOPE * a;     // leaky_relu(0.2)
    abuf[e] = a;
    atomicMax(&amaxU[d], f2ord(a));
}

// ---------------- kernel 4: exp(a - max) + segment sum ----------------------
__global__ void k_edge_exp(const int* __restrict__ dst,
                           const unsigned* __restrict__ amaxU,
                           float* __restrict__ abuf, float* __restrict__ denom,
                           int E, int n) {
    int e = blockIdx.x * blockDim.x + threadIdx.x;
    if (e >= E + n) return;
    int d = (e < E) ? dst[e] : (e - E);
    float ea = expf(abuf[e] - ord2f(amaxU[d]));
    abuf[e] = ea;
    atomicAdd(&denom[d], ea);
}

// ---------------- kernel 5: message aggregation (heavy pass) ----------------
// 4 edges per block of 256; 64 consecutive lanes cover one 256B channel row.
__global__ __launch_bounds__(256) void k_aggregate(const int* __restrict__ src,
                                                   const int* __restrict__ dst,
                                                   const float* __restrict__ abuf,
                                                   const float* __restrict__ denom,
                                                   const float* __restrict__ xl,
                                                   float* __restrict__ out,
                                                   int E, int n) {
    int c   = threadIdx.x & 63;
    int sub = threadIdx.x >> 6;
    int e   = blockIdx.x * 4 + sub;
    if (e >= E + n) return;
    int s = (e < E) ? src[e] : (e - E);
    int d = (e < E) ? dst[e] : (e - E);
    float alpha = abuf[e] / (denom[d] + 1e-16f);
    atomicAdd(&out[(size_t)d * C64 + c], alpha * xl[(size_t)s * C64 + c]);
}

// ---------------- kernel 6: BN channel sums ---------------------------------
__global__ __launch_bounds__(256) void k_bn_stats(const float* __restrict__ out,
                                                  const float* __restrict__ bias,
                                                  float* __restrict__ sums, int n) {
    int c   = threadIdx.x & 63;
    int sub = threadIdx.x >> 6;
    float b = bias[c];
    float s = 0.0f, ss = 0.0f;
    for (int r = blockIdx.x * 4 + sub; r < n; r += gridDim.x * 4) {
        float v = out[(size_t)r * C64 + c] + b;
        s += v; ss += v * v;
    }
    atomicAdd(&sums[c], s);
    atomicAdd(&sums[64 + c], ss);
}

// ---------------- kernel 7: BN finalize (mu, scale, shift) ------------------
__global__ void k_bn_final(const float* __restrict__ sums,
                           const float* __restrict__ gamma,
                           const float* __restrict__ beta,
                           float* __restrict__ params, int n) {
    int c = threadIdx.x;
    if (c >= C64) return;
    float inv_n = 1.0f / (float)n;
    float mu  = sums[c] * inv_n;
    float var = sums[64 + c] * inv_n - mu * mu;
    float rstd = rsqrtf(var + BN_EPS);
    params[c]       = mu;
    params[64 + c]  = rstd * gamma[c];
    params[128 + c] = beta[c];
}

// ---------------- kernel 8: apply BN + ReLU in place on d_out ---------------
__global__ void k_apply(float* __restrict__ out, const float* __restrict__ bias,
                        const float* __restrict__ params, int n) {
    int i = blockIdx.x * blockDim.x + threadIdx.x;
    if (i >= n * C64) return;
    int c = i & 63;
    float v = out[i] + bias[c];
    float y = (v - params[c]) * params[64 + c] + params[128 + c];
    out[i] = fmaxf(y, 0.0f);
}

extern "C" void kernel_launch(void* const* d_in, const int* in_sizes, int n_in,
                              void* d_out, int out_size, void* d_ws, size_t ws_size,
                              hipStream_t stream) {
    (void)n_in; (void)out_size; (void)ws_size;
    const float* x        = (const float*)d_in[0];
    const int*   ei       = (const int*)d_in[1];
    const float* emb      = (const float*)d_in[2];
    const float* W        = (const float*)d_in[3];
    const float* att_i    = (const float*)d_in[4];
    const float* att_j    = (const float*)d_in[5];
    const float* att_em_i = (const float*)d_in[6];
    const float* att_em_j = (const float*)d_in[7];
    const float* bias     = (const float*)d_in[8];
    const float* gamma    = (const float*)d_in[9];
    const float* beta     = (const float*)d_in[10];

    const int n = in_sizes[0] / C64;        // 50000
    const int E = in_sizes[1] / 2;          // 800000
    const int Etot = E + n;                 // + self-loops
    const int* src = ei;                    // row 0 = j (source)
    const int* dst = ei + E;                // row 1 = i (target)

    float* out = (float*)d_out;

    // workspace layout
    float*    ws    = (float*)d_ws;
    size_t    off   = 0;
    float*    xl    = ws + off;  off += (size_t)n * C64;   // [N,64]
    float*    ai    = ws + off;  off += n;
    float*    aj    = ws + off;  off += n;
    unsigned* amaxU = (unsigned*)(ws + off); off += n;
    float*    denom = ws + off;  off += n;
    float*    abuf  = ws + off;  off += Etot;              // a, then exp(a)
    float*    sums  = ws + off;  off += 128;               // sum, sumsq
    float*    params= ws + off;  off += 192;               // mu, scale, shift

    const int nc = n * C64;

    k_init<<<(nc + 255) / 256, 256, 0, stream>>>(out, denom, amaxU, sums, n);

    k_gemm_xl<<<(n + 63) / 64, 128, 0, stream>>>(x, W, xl, n);

    k_node_att<<<(n + 7) / 8, 256, 0, stream>>>(xl, emb, att_i, att_j,
                                                att_em_i, att_em_j, ai, aj, n);

    k_edge_logit<<<(Etot + 255) / 256, 256, 0, stream>>>(src, dst, ai, aj,
                                                         abuf, amaxU, E, n);

    k_edge_exp<<<(Etot + 255) / 256, 256, 0, stream>>>(dst, amaxU, abuf, denom, E, n);

    k_aggregate<<<(Etot + 3) / 4, 256, 0, stream>>>(src, dst, abuf, denom, xl,
                                                    out, E, n);

    k_bn_stats<<<256, 256, 0, stream>>>(out, bias, sums, n);

    k_bn_final<<<1, 64, 0, stream>>>(sums, gamma, beta, params, n);

    k_apply<<<(nc + 255) / 256, 256, 0, stream>>>(out, bias, params, n);
}